// HessianOut_13185549598891
// MI455X (gfx1250) — compile-verified
//
#include <hip/hip_runtime.h>
#include <math.h>

#define NN        1500
#define EDGES     (NN * NN)
#define NODE_DIM  128
#define HIDDEN    64
#define SPH_DIM   480
#define OFF0      128
#define MUL1E     64
#define HID1E     32
#define NUM_BASIS 20
#define CUTOFF    5.0f

typedef float v2f __attribute__((ext_vector_type(2)));
typedef float v8f __attribute__((ext_vector_type(8)));

// ---------------------------------------------------------------------------
// Kernel 1: per-node stage.
//   a[n][3] = perm(sph_mlp_i(x)) * (1 + scalar_mlp_i(x))
//   b[n][3] = perm(sph_mlp_j(x)) * (1 + scalar_mlp_j(x))
// One wave (32 lanes) per 16-node tile. The x@W1 [16x128]x[128x64] GEMMs for
// both scalar MLPs run on V_WMMA_F32_16X16X4_F32 (f32 matrix path, wave32).
// ---------------------------------------------------------------------------
__global__ __launch_bounds__(32)
void node_stage_kernel(const float* __restrict__ x_scalar,
                       const float* __restrict__ x_sph,
                       const float* __restrict__ Wsi1, const float* __restrict__ Wsi2,
                       const float* __restrict__ Wsj1, const float* __restrict__ Wsj2,
                       const float* __restrict__ Wpi1, const float* __restrict__ Wpi2,
                       const float* __restrict__ Wpj1, const float* __restrict__ Wpj2,
                       float* __restrict__ outA, float* __restrict__ outB)
{
    __shared__ float lds_h[2][16][HIDDEN];   // post-GEMM hidden activations
    __shared__ float acc_sph[2][16][3];      // spherical MLP accumulators
    __shared__ float s_scal[2][16];          // scalar-MLP outputs

    const int lane = threadIdx.x;            // wave32
    const int tile = blockIdx.x;             // 16 nodes per tile

    // zero spherical accumulators (96 floats)
    for (int p = lane; p < 2 * 16 * 3; p += 32)
        ((float*)acc_sph)[p] = 0.0f;

    // ---------------- scalar MLPs via WMMA f32 16x16x4 ----------------
    // A fragment (16x4 f32): lane<16 holds row M=lane, K = k0,k0+1;
    //                        lane>=16 holds row M=lane-16, K = k0+2,k0+3.
    // B fragment (4x16 f32): lane<16 holds row K=k0   (N=lane),  K=k0+1 in v1;
    //                        lane>=16 holds row K=k0+2 (N=lane-16), K=k0+3.
    // C/D (16x16 f32): vgpr r = row (r | r+8), lane&15 = column.
    const int arow   = lane & 15;
    const int khalf  = (lane >> 4) * 2;
    const int node_a = min(tile * 16 + arow, NN - 1);
    const float* xrow = x_scalar + (size_t)node_a * NODE_DIM;

    v8f acc_i[4], acc_j[4];
    #pragma unroll
    for (int t = 0; t < 4; ++t) { acc_i[t] = (v8f)(0.0f); acc_j[t] = (v8f)(0.0f); }

    for (int kk = 0; kk < NODE_DIM / 4; ++kk) {
        const int k0 = kk * 4 + khalf;
        v2f afrag;
        afrag.x = xrow[k0];
        afrag.y = xrow[k0 + 1];
        #pragma unroll
        for (int t = 0; t < 4; ++t) {
            const int col = t * 16 + (lane & 15);
            v2f bi, bj;
            bi.x = Wsi1[(size_t)k0 * HIDDEN + col];
            bi.y = Wsi1[(size_t)(k0 + 1) * HIDDEN + col];
            bj.x = Wsj1[(size_t)k0 * HIDDEN + col];
            bj.y = Wsj1[(size_t)(k0 + 1) * HIDDEN + col];
            acc_i[t] = __builtin_amdgcn_wmma_f32_16x16x4_f32(
                false, afrag, false, bi, (short)0, acc_i[t], false, false);
            acc_j[t] = __builtin_amdgcn_wmma_f32_16x16x4_f32(
                false, afrag, false, bj, (short)0, acc_j[t], false, false);
        }
    }

    // spill accumulators to LDS in (node, hidden) layout
    {
        const int mbase = (lane >> 4) * 8;
        #pragma unroll
        for (int t = 0; t < 4; ++t) {
            const int col = t * 16 + (lane & 15);
            #pragma unroll
            for (int r = 0; r < 8; ++r) {
                lds_h[0][mbase + r][col] = acc_i[t][r];
                lds_h[1][mbase + r][col] = acc_j[t][r];
            }
        }
    }
    __syncthreads();

    // finish scalar MLPs: silu then dot with W2.  lane<16 -> mlp i, else mlp j
    {
        const int m    = lane >> 4;
        const int node = lane & 15;
        const float* W2 = (m == 0) ? Wsi2 : Wsj2;
        float s = 0.0f;
        #pragma unroll 4
        for (int c = 0; c < HIDDEN; ++c) {
            const float h = lds_h[m][node][c];
            s += (h / (1.0f + __expf(-h))) * W2[c];
        }
        s_scal[m][node] = s;
    }
    __syncthreads();

    // ---------------- spherical MLPs (64x1e -> 32x1e, sigmoid gate) --------
    const float inv8      = 0.125f;            // 1/sqrt(64)
    for (int m = 0; m < 2; ++m) {
        const float* W1 = (m == 0) ? Wpi1 : Wpj1;   // [64,32]
        const float* W2 = (m == 0) ? Wpi2 : Wpj2;   // [32,1]
        for (int p = lane; p < 16 * HID1E; p += 32) {
            const int node = p >> 5;
            const int hid  = p & 31;
            const int ng   = min(tile * 16 + node, NN - 1);
            const float* v = x_sph + (size_t)ng * SPH_DIM + OFF0;
            float hx = 0.0f, hy = 0.0f, hz = 0.0f;
            #pragma unroll 8
            for (int mm = 0; mm < MUL1E; ++mm) {
                const float w = W1[mm * HID1E + hid];
                hx += v[mm * 3 + 0] * w;
                hy += v[mm * 3 + 1] * w;
                hz += v[mm * 3 + 2] * w;
            }
            hx *= inv8; hy *= inv8; hz *= inv8;
            const float nrm  = sqrtf(hx * hx + hy * hy + hz * hz);
            const float gate = 1.0f / (1.0f + __expf(-nrm));
            const float cf   = gate * W2[hid];
            atomicAdd(&acc_sph[m][node][0], hx * cf);
            atomicAdd(&acc_sph[m][node][1], hy * cf);
            atomicAdd(&acc_sph[m][node][2], hz * cf);
        }
    }
    __syncthreads();

    // combine:  perm (y,z,x)->(x,y,z)  i.e.  out[d] = raw[(2,0,1)[d]]
    {
        const int m    = lane >> 4;
        const int node = lane & 15;
        const int ng   = tile * 16 + node;
        if (ng < NN) {
            const float invs32 = 0.17677669529663688f;  // 1/sqrt(32)
            const float g = 1.0f + s_scal[m][node];
            const float r0 = acc_sph[m][node][0] * invs32;
            const float r1 = acc_sph[m][node][1] * invs32;
            const float r2 = acc_sph[m][node][2] * invs32;
            float* dst = ((m == 0) ? outA : outB) + (size_t)ng * 3;
            dst[0] = r2 * g;
            dst[1] = r0 * g;
            dst[2] = r1 * g;
        }
    }
}

// ---------------------------------------------------------------------------
// Kernel 2: edge stage (write-bandwidth bound, 81 MB streaming output).
//   out[i*N+j][r][c] = 0.5 * s(d_ij) * (a_i[r]*b_j[c] + b_i[r]*a_j[c])
// One thread per edge; 9 contiguous dwords per thread with NT store hints.
// ---------------------------------------------------------------------------
__global__ __launch_bounds__(256)
void edge_stage_kernel(const float* __restrict__ A,
                       const float* __restrict__ B,
                       const float* __restrict__ coord,
                       const float* __restrict__ Wrbf,
                       float* __restrict__ out)
{
    const unsigned e = blockIdx.x * 256u + threadIdx.x;
    if (e >= (unsigned)EDGES) return;
    const unsigned i = e / (unsigned)NN;
    const unsigned j = e - i * (unsigned)NN;

    const float dx = coord[i * 3 + 0] - coord[j * 3 + 0];
    const float dy = coord[i * 3 + 1] - coord[j * 3 + 1];
    const float dz = coord[i * 3 + 2] - coord[j * 3 + 2];
    const float d  = sqrtf(dx * dx + dy * dy + dz * dz);

    // fc_rbf: 20 gaussians dotted with Wrbf (L1-resident constants)
    const float delta = CUTOFF / (float)(NUM_BASIS - 1);
    const float coeff = -0.5f / (delta * delta);
    float s = 0.0f;
    #pragma unroll
    for (int k = 0; k < NUM_BASIS; ++k) {
        const float t = d - (float)k * delta;
        s += Wrbf[k] * __expf(coeff * t * t);
    }
    s *= 0.5f;

    const float ai0 = A[i * 3 + 0], ai1 = A[i * 3 + 1], ai2 = A[i * 3 + 2];
    const float bi0 = B[i * 3 + 0], bi1 = B[i * 3 + 1], bi2 = B[i * 3 + 2];
    const float aj0 = A[j * 3 + 0], aj1 = A[j * 3 + 1], aj2 = A[j * 3 + 2];
    const float bj0 = B[j * 3 + 0], bj1 = B[j * 3 + 1], bj2 = B[j * 3 + 2];

    float* o = out + (size_t)e * 9;
    __builtin_nontemporal_store(s * (ai0 * bj0 + bi0 * aj0), o + 0);
    __builtin_nontemporal_store(s * (ai0 * bj1 + bi0 * aj1), o + 1);
    __builtin_nontemporal_store(s * (ai0 * bj2 + bi0 * aj2), o + 2);
    __builtin_nontemporal_store(s * (ai1 * bj0 + bi1 * aj0), o + 3);
    __builtin_nontemporal_store(s * (ai1 * bj1 + bi1 * aj1), o + 4);
    __builtin_nontemporal_store(s * (ai1 * bj2 + bi1 * aj2), o + 5);
    __builtin_nontemporal_store(s * (ai2 * bj0 + bi2 * aj0), o + 6);
    __builtin_nontemporal_store(s * (ai2 * bj1 + bi2 * aj1), o + 7);
    __builtin_nontemporal_store(s * (ai2 * bj2 + bi2 * aj2), o + 8);
}

extern "C" void kernel_launch(void* const* d_in, const int* in_sizes, int n_in,
                              void* d_out, int out_size, void* d_ws, size_t ws_size,
                              hipStream_t stream) {
    const float* x_scalar = (const float*)d_in[0];
    const float* x_sph    = (const float*)d_in[1];
    const float* coord    = (const float*)d_in[2];
    // d_in[3] = fc_edge_index (implicit full graph; not needed)
    const float* Wsi1 = (const float*)d_in[4];
    const float* Wsi2 = (const float*)d_in[5];
    const float* Wsj1 = (const float*)d_in[6];
    const float* Wsj2 = (const float*)d_in[7];
    const float* Wpi1 = (const float*)d_in[8];
    const float* Wpi2 = (const float*)d_in[9];
    const float* Wpj1 = (const float*)d_in[10];
    const float* Wpj2 = (const float*)d_in[11];
    const float* Wrbf = (const float*)d_in[12];

    float* A = (float*)d_ws;             // [N,3]
    float* B = A + (size_t)NN * 3;       // [N,3]
    float* out = (float*)d_out;          // [E,3,3]

    const int tiles = (NN + 15) / 16;
    node_stage_kernel<<<tiles, 32, 0, stream>>>(
        x_scalar, x_sph, Wsi1, Wsi2, Wsj1, Wsj2, Wpi1, Wpi2, Wpj1, Wpj2, A, B);

    const int blocks = (EDGES + 255) / 256;
    edge_stage_kernel<<<blocks, 256, 0, stream>>>(A, B, coord, Wrbf, out);
}